// Adder2D_1795296330278
// MI455X (gfx1250) — compile-verified
//
#include <hip/hip_runtime.h>
#include <hip/hip_bf16.h>
#include <stdint.h>

// ---- feature probes -------------------------------------------------------
#if defined(__has_builtin)
#if __has_builtin(__builtin_amdgcn_global_load_async_to_lds_b128)
#define HAVE_ASYNC_B128 1
#endif
#if __has_builtin(__builtin_amdgcn_s_wait_asynccnt)
#define HAVE_WAIT_ASYNCCNT 1
#endif
#if __has_builtin(__builtin_amdgcn_wmma_f32_16x16x4_f32)
#define HAVE_WMMA_F32_16X16X4 1
#endif
#endif

typedef int v4i __attribute__((vector_size(16)));   // matches builtin pointee
typedef __attribute__((address_space(1))) v4i glb_v4i;
typedef __attribute__((address_space(3))) v4i lds_v4i;
typedef __attribute__((ext_vector_type(2))) float v2f;
typedef __attribute__((ext_vector_type(8))) float v8f;

// 16B global -> LDS copy; async CDNA5 path, sync fallback otherwise.
__device__ __forceinline__ void copy16_g2l(const float* g, float* l) {
#if defined(HAVE_ASYNC_B128)
  // integer detours avoid address-space conversion pitfalls:
  // AS1 pointer is 64-bit; AS3 pointer is the low-32-bit LDS offset.
  __builtin_amdgcn_global_load_async_to_lds_b128(
      (glb_v4i*)(uintptr_t)g,
      (lds_v4i*)(uint32_t)(uintptr_t)l,
      0, 0);
#else
  *(float4*)l = *(const float4*)g;
#endif
}

__device__ __forceinline__ void wait_async_copies() {
#if defined(HAVE_ASYNC_B128)
#if defined(HAVE_WAIT_ASYNCCNT)
  __builtin_amdgcn_s_wait_asynccnt(0);
#else
  asm volatile("s_wait_asynccnt 0" ::: "memory");
#endif
#endif
}

// x: [4,64,64,32] f32 NHWC, kernel: [3,3,32,32] f32 (ky,kx,ci,co)
// out[n,h,w,co] = -sum_{ky,kx,ci} | x_pad[n,h+ky-1,w+kx-1,ci] - k[ky,kx,ci,co] |
// One workgroup (8 waves) per (n,h) row. lane = co (wave32 == Cout).
// Each wave owns 8 consecutive w positions.
__global__ __launch_bounds__(256) void adder2d_kernel(const float* __restrict__ x,
                                                      const float* __restrict__ kern,
                                                      float* __restrict__ out) {
  struct SmemT {
    float w[9 * 32 * 32];   // [t=(ky*3+kx)][ci][co]   36 KB
    float xr[3][64 * 32];   // three staged input rows 24 KB
    float absw[9 * 32];     // sum_ci |w[t][ci][co]|   1.1 KB
  };
  __shared__ SmemT S;

  const int tid  = (int)threadIdx.x;
  const int lane = tid & 31;   // co
  const int wv   = tid >> 5;   // wave id 0..7
  const int bid  = (int)blockIdx.x;   // n*64 + h
  const int h    = bid & 63;

  // ---- stage weights: 9216 floats = 2304 x b128, 9 per thread -------------
#pragma unroll
  for (int i = 0; i < 9; ++i) {
    const int c = tid + i * 256;
    copy16_g2l(kern + c * 4, S.w + c * 4);
  }
  // ---- stage 3 input rows (2048 floats each = 512 x b128, 2 per thread) ---
#pragma unroll
  for (int r = 0; r < 3; ++r) {
    const int hr = h - 1 + r;
    if ((unsigned)hr < 64u) {
      const float* src = x + (size_t)(bid - 1 + r) * 2048;  // (n*64+hr)*64*32
#pragma unroll
      for (int i = 0; i < 2; ++i) {
        const int c = tid + i * 256;
        copy16_g2l(src + c * 4, S.xr[r] + c * 4);
      }
    }
  }
  wait_async_copies();
  __syncthreads();

  // ---- absw[t][co] = sum_ci |w[t][ci][co]|  (padded-tap contribution) -----
#if defined(HAVE_WMMA_F32_16X16X4)
  // ones(16x4) x |W|(4x16) accumulated over 8 k-chunks: every D row = col sums.
  // B layout (4x16 f32): lanes 0-15 hold K=0(V0),K=1(V1); lanes 16-31 K=2,K=3.
  for (int q = wv; q < 18; q += 8) {          // 9 taps x 2 co-halves
    const int t  = q >> 1;
    const int hf = q & 1;
    const int co = (lane & 15) + hf * 16;
    const int k0 = (lane < 16) ? 0 : 2;
    v8f acc = {};
    v2f a;  a.x = 1.0f;  a.y = 1.0f;          // A = ones (16x4)
#pragma unroll
    for (int c8 = 0; c8 < 8; ++c8) {
      const int ci0 = c8 * 4 + k0;
      v2f b;
      b.x = __builtin_fabsf(S.w[(t * 32 + ci0) * 32 + co]);
      b.y = __builtin_fabsf(S.w[(t * 32 + ci0 + 1) * 32 + co]);
      acc = __builtin_amdgcn_wmma_f32_16x16x4_f32(false, a, false, b,
                                                  (short)0, acc, false, false);
    }
    if (lane < 16) S.absw[t * 32 + hf * 16 + lane] = acc[0];  // D row M=0
  }
#else
  for (int idx = tid; idx < 288; idx += 256) {
    const int t = idx >> 5, co = idx & 31;
    float s = 0.0f;
#pragma unroll
    for (int ci = 0; ci < 32; ++ci) s += __builtin_fabsf(S.w[(t * 32 + ci) * 32 + co]);
    S.absw[idx] = s;
  }
#endif
  __syncthreads();

  // ---- main loop: lane=co, 8 pixels per wave ------------------------------
  float acc[8];
#pragma unroll
  for (int p = 0; p < 8; ++p) acc[p] = 0.0f;
  const int wbase = wv * 8;

#pragma unroll
  for (int ky = 0; ky < 3; ++ky) {
    const bool rowok = ((unsigned)(h - 1 + ky) < 64u);
#pragma unroll
    for (int kx = 0; kx < 3; ++kx) {
      const int t = ky * 3 + kx;
      // per-lane weights for this tap: conflict-free lane-consecutive ds loads
      float wreg[32];
      const float* wp = S.w + t * 1024 + lane;
#pragma unroll
      for (int ci = 0; ci < 32; ++ci) wreg[ci] = wp[ci * 32];
      const float aw = S.absw[t * 32 + lane];

#pragma unroll
      for (int p = 0; p < 8; ++p) {
        const int col = wbase + p - 1 + kx;
        if (rowok && (unsigned)col < 64u) {
          // uniform-address LDS reads -> hardware broadcast to all 32 lanes
          const float4* xq = (const float4*)(S.xr[ky] + col * 32);
#pragma unroll
          for (int c4 = 0; c4 < 8; ++c4) {
            const float4 xv = xq[c4];
            acc[p] -= __builtin_fabsf(xv.x - wreg[c4 * 4 + 0]);
            acc[p] -= __builtin_fabsf(xv.y - wreg[c4 * 4 + 1]);
            acc[p] -= __builtin_fabsf(xv.z - wreg[c4 * 4 + 2]);
            acc[p] -= __builtin_fabsf(xv.w - wreg[c4 * 4 + 3]);
          }
        } else {
          acc[p] -= aw;   // zero-padded tap contributes sum|w|
        }
      }
    }
  }

  // ---- store: lanes write consecutive co -> fully coalesced ---------------
  float* op = out + (size_t)bid * 64 * 32 + lane;
#pragma unroll
  for (int p = 0; p < 8; ++p) op[(size_t)(wbase + p) * 32] = acc[p];
}

extern "C" void kernel_launch(void* const* d_in, const int* in_sizes, int n_in,
                              void* d_out, int out_size, void* d_ws, size_t ws_size,
                              hipStream_t stream) {
  (void)in_sizes; (void)n_in; (void)out_size; (void)d_ws; (void)ws_size;
  const float* x  = (const float*)d_in[0];   // [4,64,64,32]
  const float* kn = (const float*)d_in[1];   // [3,3,32,32]
  float* out = (float*)d_out;                // [4,64,64,32]
  adder2d_kernel<<<dim3(4 * 64), dim3(256), 0, stream>>>(x, kn, out);
}